// HexPlaneField_84937273245764
// MI455X (gfx1250) — compile-verified
//
#include <hip/hip_runtime.h>
#include <stdint.h>

// ---------------------------------------------------------------------------
// HexPlane forward for MI455X (gfx1250).
//  Pass 1: transpose each grid (32,H,W) -> (H,W,32) in d_ws using CDNA5
//          async gather-to-LDS (global_load_async_to_lds_b32 / s_wait_asynccnt).
//  Pass 2: bilinear sample with channel-last layout: coalesced b128 gathers
//          from the L2-resident (46MB < 192MB L2) transposed grids,
//          non-temporal output stores to keep grids hot in L2.
//  All gather offsets kept in 32-bit so the compiler can use SGPR-base +
//  32-bit VGPR-offset (GVS) addressing instead of 64-bit VALU address math.
// ---------------------------------------------------------------------------

typedef __attribute__((ext_vector_type(4))) float v4f;

struct GridPtrs { const float* p[18]; };

__device__ __forceinline__ v4f ld4(const float* g, int off) {
    return *(const v4f*)(g + off);
}

// ---------------- Pass 1: (C=32,H,W) -> (H,W,C=32) transpose -----------------
__global__ void __launch_bounds__(256)
transpose_chw_to_hwc(const float* __restrict__ src, float* __restrict__ dst,
                     int H, int W)
{
    __shared__ float sm[32 * 33];   // +1 pad: conflict-free both phases
    const int tid   = threadIdx.x;
    const int xBase = blockIdx.x * 32;
    const int y     = blockIdx.y;
    const int x     = tid & 31;     // lane id within wave32 == x
    const int cq    = tid >> 5;     // 0..7

#pragma unroll
    for (int r = 0; r < 4; ++r) {
        const int c = cq * 4 + r;
        const float* g = src + ((size_t)c * H + y) * W + xBase + x;
        const uint32_t lds_off = (uint32_t)(uintptr_t)(&sm[x * 33 + c]);
        // CDNA5 async copy: copies a float from global address g to LDS offset lds_off; tracked by ASYNCcnt.
        asm volatile("global_load_async_to_lds_b32 %0, %1, off"
                     :: "v"(lds_off), "v"((uint64_t)(uintptr_t)g)
                     : "memory");
    }
#if __has_builtin(__builtin_amdgcn_s_wait_asynccnt)
    __builtin_amdgcn_s_wait_asynccnt(0);
#else
    asm volatile("s_wait_asynccnt 0x0" ::: "memory");
#endif
    __syncthreads();

#pragma unroll
    for (int r = 0; r < 4; ++r) {
        const int idx = tid + r * 256;
        const int c   = idx & 31;
        const int xx  = idx >> 5;
        dst[((size_t)y * W + xBase + xx) * 32 + c] = sm[xx * 33 + c];
    }
}

// ---------------- Pass 2: bilinear sampler, channel-last ---------------------
// thread = (point, 8-channel group); 4 threads cover 32 channels of a point.
__global__ void __launch_bounds__(256)
hexplane_sample_cl(const float* __restrict__ pts, const float* __restrict__ tsv,
                   GridPtrs G, float* __restrict__ out, int N)
{
    const int t  = blockIdx.x * 256 + threadIdx.x;
    const int p  = t >> 2;
    const int cg = t & 3;
    if (p >= N) return;

    const float px = __builtin_nontemporal_load(pts + 3 * (size_t)p + 0);
    const float py = __builtin_nontemporal_load(pts + 3 * (size_t)p + 1);
    const float pz = __builtin_nontemporal_load(pts + 3 * (size_t)p + 2);
    const float tv = __builtin_nontemporal_load(tsv + (size_t)p);

    float P[4];
    P[0] = (px - 1.6f) * (-0.625f) - 1.0f;   // 2/(-2*1.6) = -0.625
    P[1] = (py - 1.6f) * (-0.625f) - 1.0f;
    P[2] = (pz - 1.6f) * (-0.625f) - 1.0f;
    P[3] = (2.0f * tv * 75.0f) / 74.0f - 1.0f;

    const int cb = cg * 8;
    const int I[6] = {0, 0, 0, 1, 1, 2};
    const int J[6] = {1, 2, 3, 2, 3, 3};

#pragma unroll
    for (int s = 0; s < 3; ++s) {
        const int R = 64 << s;
        float acc[8] = {0.f, 0.f, 0.f, 0.f, 0.f, 0.f, 0.f, 0.f};
#pragma unroll
        for (int k = 0; k < 6; ++k) {
            const int W = R;
            const int H = (J[k] == 3) ? 75 : R;
            const float cx = P[I[k]];
            const float cy = P[J[k]];
            float ix = (cx + 1.0f) * 0.5f * (float)(W - 1);
            ix = fminf(fmaxf(ix, 0.0f), (float)(W - 1));
            float iy = (cy + 1.0f) * 0.5f * (float)(H - 1);
            iy = fminf(fmaxf(iy, 0.0f), (float)(H - 1));
            const float x0f = floorf(ix), y0f = floorf(iy);
            const float wx = ix - x0f,   wy = iy - y0f;
            const int x0 = (int)x0f, y0 = (int)y0f;
            const int x1 = min(x0 + 1, W - 1);
            const int y1 = min(y0 + 1, H - 1);
            const float u = 1.0f - wx, v = 1.0f - wy;
            const float w00 = u * v,  w10 = wx * v;
            const float w01 = u * wy, w11 = wx * wy;

            // 32-bit element offsets (max plane = 2M floats, fits easily)
            const int r0  = y0 * W;
            const int r1  = y1 * W;
            const int o00 = (r0 + x0) * 32 + cb;
            const int o10 = (r0 + x1) * 32 + cb;
            const int o01 = (r1 + x0) * 32 + cb;
            const int o11 = (r1 + x1) * 32 + cb;
            const float* g = G.p[s * 6 + k];

            v4f q;
            q = ld4(g, o00);     acc[0] += w00*q[0]; acc[1] += w00*q[1]; acc[2] += w00*q[2]; acc[3] += w00*q[3];
            q = ld4(g, o00 + 4); acc[4] += w00*q[0]; acc[5] += w00*q[1]; acc[6] += w00*q[2]; acc[7] += w00*q[3];
            q = ld4(g, o10);     acc[0] += w10*q[0]; acc[1] += w10*q[1]; acc[2] += w10*q[2]; acc[3] += w10*q[3];
            q = ld4(g, o10 + 4); acc[4] += w10*q[0]; acc[5] += w10*q[1]; acc[6] += w10*q[2]; acc[7] += w10*q[3];
            q = ld4(g, o01);     acc[0] += w01*q[0]; acc[1] += w01*q[1]; acc[2] += w01*q[2]; acc[3] += w01*q[3];
            q = ld4(g, o01 + 4); acc[4] += w01*q[0]; acc[5] += w01*q[1]; acc[6] += w01*q[2]; acc[7] += w01*q[3];
            q = ld4(g, o11);     acc[0] += w11*q[0]; acc[1] += w11*q[1]; acc[2] += w11*q[2]; acc[3] += w11*q[3];
            q = ld4(g, o11 + 4); acc[4] += w11*q[0]; acc[5] += w11*q[1]; acc[6] += w11*q[2]; acc[7] += w11*q[3];
        }
        float* ob = out + (size_t)p * 96 + s * 32 + cb;
        v4f o0 = { acc[0], acc[1], acc[2], acc[3] };
        v4f o1 = { acc[4], acc[5], acc[6], acc[7] };
        __builtin_nontemporal_store(o0, (v4f*)ob);
        __builtin_nontemporal_store(o1, (v4f*)(ob + 4));
    }
}

// ---------------- Fallback: direct (C,H,W) layout sampler --------------------
__global__ void __launch_bounds__(256)
hexplane_sample_direct(const float* __restrict__ pts, const float* __restrict__ tsv,
                       GridPtrs G, float* __restrict__ out, int N)
{
    const int t  = blockIdx.x * 256 + threadIdx.x;
    const int p  = t >> 2;
    const int cg = t & 3;
    if (p >= N) return;

    const float px = pts[3 * (size_t)p + 0];
    const float py = pts[3 * (size_t)p + 1];
    const float pz = pts[3 * (size_t)p + 2];
    const float tv = tsv[(size_t)p];

    float P[4];
    P[0] = (px - 1.6f) * (-0.625f) - 1.0f;
    P[1] = (py - 1.6f) * (-0.625f) - 1.0f;
    P[2] = (pz - 1.6f) * (-0.625f) - 1.0f;
    P[3] = (2.0f * tv * 75.0f) / 74.0f - 1.0f;

    const int cb = cg * 8;
    const int I[6] = {0, 0, 0, 1, 1, 2};
    const int J[6] = {1, 2, 3, 2, 3, 3};

#pragma unroll
    for (int s = 0; s < 3; ++s) {
        const int R = 64 << s;
        float acc[8] = {0.f, 0.f, 0.f, 0.f, 0.f, 0.f, 0.f, 0.f};
#pragma unroll
        for (int k = 0; k < 6; ++k) {
            const int W = R;
            const int H = (J[k] == 3) ? 75 : R;
            const float cx = P[I[k]];
            const float cy = P[J[k]];
            float ix = (cx + 1.0f) * 0.5f * (float)(W - 1);
            ix = fminf(fmaxf(ix, 0.0f), (float)(W - 1));
            float iy = (cy + 1.0f) * 0.5f * (float)(H - 1);
            iy = fminf(fmaxf(iy, 0.0f), (float)(H - 1));
            const float x0f = floorf(ix), y0f = floorf(iy);
            const float wx = ix - x0f,   wy = iy - y0f;
            const int x0 = (int)x0f, y0 = (int)y0f;
            const int x1 = min(x0 + 1, W - 1);
            const int y1 = min(y0 + 1, H - 1);
            const float u = 1.0f - wx, v = 1.0f - wy;
            const float w00 = u * v,  w10 = wx * v;
            const float w01 = u * wy, w11 = wx * wy;

            const int hw  = H * W;
            const float* g = G.p[s * 6 + k] + cb * hw;
            const int o00 = y0 * W + x0;
            const int o10 = y0 * W + x1;
            const int o01 = y1 * W + x0;
            const int o11 = y1 * W + x1;
#pragma unroll
            for (int c = 0; c < 8; ++c) {
                const float* gc = g + c * hw;
                acc[c] += w00 * gc[o00] + w10 * gc[o10] + w01 * gc[o01] + w11 * gc[o11];
            }
        }
        float* ob = out + (size_t)p * 96 + s * 32 + cb;
        v4f o0 = { acc[0], acc[1], acc[2], acc[3] };
        v4f o1 = { acc[4], acc[5], acc[6], acc[7] };
        __builtin_nontemporal_store(o0, (v4f*)ob);
        __builtin_nontemporal_store(o1, (v4f*)(ob + 4));
    }
}

// ---------------------------------------------------------------------------
extern "C" void kernel_launch(void* const* d_in, const int* in_sizes, int n_in,
                              void* d_out, int out_size, void* d_ws, size_t ws_size,
                              hipStream_t stream)
{
    const float* pts = (const float*)d_in[0];
    const float* tsv = (const float*)d_in[1];
    const int N = in_sizes[0] / 3;

    static const int Jt[6] = {1, 2, 3, 2, 3, 3};

    size_t elems[18];
    size_t total = 0;
    for (int k = 0; k < 18; ++k) {
        const int s = k / 6, kk = k % 6, R = 64 << s;
        const int W = R, H = (Jt[kk] == 3) ? 75 : R;
        elems[k] = (size_t)32 * H * W;
        total += elems[k];
    }

    const float* gsrc[18];
    if (n_in >= 20) {
        for (int k = 0; k < 18; ++k) gsrc[k] = (const float*)d_in[2 + k];
    } else {
        // grids delivered as one flat concatenation (pytree leaf order)
        const float* b = (const float*)d_in[2];
        size_t off = 0;
        for (int k = 0; k < 18; ++k) { gsrc[k] = b + off; off += elems[k]; }
    }

    float* outf = (float*)d_out;
    const bool use_ws = (ws_size >= total * sizeof(float));

    const long threads = (long)N * 4;
    const int blocks = (int)((threads + 255) / 256);

    if (use_ws) {
        float* wsf = (float*)d_ws;
        GridPtrs G;
        size_t off = 0;
        for (int k = 0; k < 18; ++k) {
            const int s = k / 6, kk = k % 6, R = 64 << s;
            const int W = R, H = (Jt[kk] == 3) ? 75 : R;
            float* dst = wsf + off;
            off += elems[k];
            dim3 grid(W / 32, H);
            transpose_chw_to_hwc<<<grid, 256, 0, stream>>>(gsrc[k], dst, H, W);
            G.p[k] = dst;
        }
        hexplane_sample_cl<<<blocks, 256, 0, stream>>>(pts, tsv, G, outf, N);
    } else {
        GridPtrs G;
        for (int k = 0; k < 18; ++k) G.p[k] = gsrc[k];
        hexplane_sample_direct<<<blocks, 256, 0, stream>>>(pts, tsv, G, outf, N);
    }
}